// TMscoreHead_57415122813285
// MI455X (gfx1250) — compile-verified
//
#include <hip/hip_runtime.h>

// TM-score head for MI455X (gfx1250, wave32).
// Memory-bound: stream P/Q twice (second pass hits 192MB L2), accumulate 16
// covariance stats per lane, reduce across the wave with V_WMMA_F32_16X16X4_F32
// (B = all-ones makes the B-layout irrelevant; A/C layouts per CDNA5 ISA 7.12.2).

#define BB 512
#define LL 2048
#define AAT 3
#define NN (LL * AAT)          // 6144 points per batch
#define SPLIT 2
#define PTS_PER_BLOCK (NN / SPLIT)   // 3072
#define RES_PER_BLOCK (LL / SPLIT)   // 1024

typedef float v2f __attribute__((ext_vector_type(2)));
typedef float v8f __attribute__((ext_vector_type(8)));

// ---------------------------------------------------------------- pass 1 ----
// Per (batch, split) block: accumulate a ⊗ b with a=(w*px,w*py,w*pz,w),
// b=(qx,qy,qz,1)  ->  acc[4i+j]. Gives M(3x3), sumP, sumQ, cnt in 16 floats.
__global__ __launch_bounds__(256) void tm_pass1(
    const float* __restrict__ P, const float* __restrict__ Q,
    const unsigned char* __restrict__ CM, const unsigned char* __restrict__ RM,
    float* __restrict__ ws1) {
  const int blk = blockIdx.x;
  const int b = blk >> 1;
  const int split = blk & 1;
  const int tid = threadIdx.x;
  const int lane = tid & 31;
  const int warp = tid >> 5;

  __shared__ float ldsT[8][4][16];   // per-wave transpose staging
  __shared__ float wavePart[8][16];  // per-wave stat totals
  __shared__ float waveN[8];

  float acc[16];
#pragma unroll
  for (int s = 0; s < 16; ++s) acc[s] = 0.f;

  const size_t pbase = (size_t)b * NN;
  const float4* __restrict__ Pv = (const float4*)(P + pbase * 3);
  const float4* __restrict__ Qv = (const float4*)(Q + pbase * 3);
  const unsigned int* __restrict__ CMv = (const unsigned int*)(CM + pbase);
  const int g0 = (split * PTS_PER_BLOCK) >> 2;  // first 4-point group

#pragma unroll
  for (int it = 0; it < 3; ++it) {
    const int p4 = g0 + it * 256 + tid;  // 4-point group index
    // prefetch next iteration's cachelines (global_prefetch_b8)
    __builtin_prefetch((const void*)(Pv + (size_t)(p4 + 256) * 3), 0, 0);
    __builtin_prefetch((const void*)(Qv + (size_t)(p4 + 256) * 3), 0, 0);
    const float4 x0 = Pv[(size_t)p4 * 3 + 0];
    const float4 x1 = Pv[(size_t)p4 * 3 + 1];
    const float4 x2 = Pv[(size_t)p4 * 3 + 2];
    const float4 y0 = Qv[(size_t)p4 * 3 + 0];
    const float4 y1 = Qv[(size_t)p4 * 3 + 1];
    const float4 y2 = Qv[(size_t)p4 * 3 + 2];
    const unsigned int m4 = CMv[p4];

    const float px[4] = {x0.x, x0.w, x1.z, x2.y};
    const float py[4] = {x0.y, x1.x, x1.w, x2.z};
    const float pz[4] = {x0.z, x1.y, x2.x, x2.w};
    const float qx[4] = {y0.x, y0.w, y1.z, y2.y};
    const float qy[4] = {y0.y, y1.x, y1.w, y2.z};
    const float qz[4] = {y0.z, y1.y, y2.x, y2.w};

#pragma unroll
    for (int k = 0; k < 4; ++k) {
      const float w = (float)((m4 >> (8 * k)) & 1u);
      const float a0 = w * px[k], a1 = w * py[k], a2 = w * pz[k];
      const float b0 = qx[k], b1 = qy[k], b2 = qz[k];
      acc[0]  = fmaf(a0, b0, acc[0]);  acc[1]  = fmaf(a0, b1, acc[1]);
      acc[2]  = fmaf(a0, b2, acc[2]);  acc[3]  += a0;
      acc[4]  = fmaf(a1, b0, acc[4]);  acc[5]  = fmaf(a1, b1, acc[5]);
      acc[6]  = fmaf(a1, b2, acc[6]);  acc[7]  += a1;
      acc[8]  = fmaf(a2, b0, acc[8]);  acc[9]  = fmaf(a2, b1, acc[9]);
      acc[10] = fmaf(a2, b2, acc[10]); acc[11] += a2;
      acc[12] = fmaf(w, b0, acc[12]);  acc[13] = fmaf(w, b1, acc[13]);
      acc[14] = fmaf(w, b2, acc[14]);  acc[15] += w;
    }
  }

  // residue count (for d0): 1024 residues / block = 256 uint loads
  const unsigned int* __restrict__ RMv =
      (const unsigned int*)(RM + (size_t)b * LL);
  const unsigned int rm = RMv[((split * RES_PER_BLOCK) >> 2) + tid];
  float nres = (float)((rm & 1u) + ((rm >> 8) & 1u) + ((rm >> 16) & 1u) +
                       ((rm >> 24) & 1u));

  // --- reduce to 4 partials per stat within each group of 8 lanes ---
#pragma unroll
  for (int m = 1; m <= 4; m <<= 1) {
#pragma unroll
    for (int s = 0; s < 16; ++s) acc[s] += __shfl_xor(acc[s], m, 32);
  }
#pragma unroll
  for (int m = 1; m <= 16; m <<= 1) nres += __shfl_xor(nres, m, 32);

  if ((lane & 7) == 0) {
    const int g = lane >> 3;
#pragma unroll
    for (int s = 0; s < 16; ++s) ldsT[warp][g][s] = acc[s];
  }
  if (lane == 0) waveN[warp] = nres;
  __syncthreads();

  // --- WMMA cross-lane reduction: D[m][*] = sum_k A[m][k] with B == 1 ---
  // A layout (16x4 f32): lanes 0-15 hold K=0 (vgpr0), K=1 (vgpr1);
  // lanes 16-31 hold K=2, K=3 for the same M=lane&15.
  {
    const int srow = lane & 15;
    const int gb = (lane >> 4) << 1;
    v2f aA;
    aA.x = ldsT[warp][gb + 0][srow];
    aA.y = ldsT[warp][gb + 1][srow];
#if __has_builtin(__builtin_amdgcn_wmma_f32_16x16x4_f32)
    v2f bB;
    bB.x = 1.0f;
    bB.y = 1.0f;
    v8f cC = {0.f, 0.f, 0.f, 0.f, 0.f, 0.f, 0.f, 0.f};
    v8f dD = __builtin_amdgcn_wmma_f32_16x16x4_f32(
        false, aA, false, bB, (short)0, cC, false, false);
    // C layout: vgpr r, lane 0 -> C[r][0]; lane 16 -> C[r+8][0]
    if (lane == 0) {
#pragma unroll
      for (int r = 0; r < 8; ++r) wavePart[warp][r] = dD[r];
    }
    if (lane == 16) {
#pragma unroll
      for (int r = 0; r < 8; ++r) wavePart[warp][8 + r] = dD[r];
    }
#else
    // Fallback (no wmma builtin): serial 4-way add per stat.
    if (lane < 16) {
      wavePart[warp][lane] = ldsT[warp][0][lane] + ldsT[warp][1][lane] +
                             ldsT[warp][2][lane] + ldsT[warp][3][lane];
    }
    (void)aA;
#endif
  }
  __syncthreads();

  if (tid < 16) {
    float t = 0.f;
#pragma unroll
    for (int w8 = 0; w8 < 8; ++w8) t += wavePart[w8][tid];
    ws1[(size_t)blk * 32 + tid] = t;
  } else if (tid == 16) {
    float t = 0.f;
#pragma unroll
    for (int w8 = 0; w8 < 8; ++w8) t += waveN[w8];
    ws1[(size_t)blk * 32 + 16] = t;
  }
}

// ---------------------------------------------------------------- kabsch ----
template <int p, int q, int r>
__device__ __forceinline__ void jrot(float K[9], float V[9]) {
  const float apq = K[p * 3 + q];
  if (fabsf(apq) < 1e-30f) return;
  const float tau = (K[q * 3 + q] - K[p * 3 + p]) * 0.5f / apq;
  const float t =
      (tau >= 0.f ? 1.f : -1.f) / (fabsf(tau) + sqrtf(1.f + tau * tau));
  const float c = rsqrtf(1.f + t * t);
  const float s = t * c;
  K[p * 3 + p] -= t * apq;
  K[q * 3 + q] += t * apq;
  K[p * 3 + q] = K[q * 3 + p] = 0.f;
  const float krp = K[r * 3 + p], krq = K[r * 3 + q];
  K[r * 3 + p] = K[p * 3 + r] = c * krp - s * krq;
  K[r * 3 + q] = K[q * 3 + r] = s * krp + c * krq;
#pragma unroll
  for (int i = 0; i < 3; ++i) {
    const float vp = V[i * 3 + p], vq = V[i * 3 + q];
    V[i * 3 + p] = c * vp - s * vq;
    V[i * 3 + q] = s * vp + c * vq;
  }
}

__global__ void tm_kabsch(const float* __restrict__ ws1,
                          float* __restrict__ ws2) {
  const int b = blockIdx.x * blockDim.x + threadIdx.x;
  if (b >= BB) return;

  float st[17];
#pragma unroll
  for (int s = 0; s < 17; ++s)
    st[s] = ws1[(size_t)(b * SPLIT + 0) * 32 + s] +
            ws1[(size_t)(b * SPLIT + 1) * 32 + s];

  const float cnt = st[15];
  const float cs = fmaxf(cnt, 1.f);
  float muP[3], muQ[3], H[9];
#pragma unroll
  for (int i = 0; i < 3; ++i) muP[i] = st[4 * i + 3] / cs;
#pragma unroll
  for (int j = 0; j < 3; ++j) muQ[j] = st[12 + j] / cs;
#pragma unroll
  for (int i = 0; i < 3; ++i)
#pragma unroll
    for (int j = 0; j < 3; ++j)
      H[i * 3 + j] = st[4 * i + j] - muP[i] * st[12 + j];

  // K = H^T H, Jacobi eigendecomposition -> V, lambda
  float K[9];
#pragma unroll
  for (int i = 0; i < 3; ++i)
#pragma unroll
    for (int j = 0; j < 3; ++j)
      K[i * 3 + j] = H[0 + i] * H[0 + j] + H[3 + i] * H[3 + j] +
                     H[6 + i] * H[6 + j];
  float V[9] = {1.f, 0.f, 0.f, 0.f, 1.f, 0.f, 0.f, 0.f, 1.f};
#pragma unroll
  for (int sweep = 0; sweep < 10; ++sweep) {
    jrot<0, 1, 2>(K, V);
    jrot<0, 2, 1>(K, V);
    jrot<1, 2, 0>(K, V);
  }
  float lam[3] = {K[0], K[4], K[8]};
  // sort eigenpairs descending (swap V columns)
#define SWAPCOL(a, c)                                                   \
  do {                                                                  \
    float tl = lam[a]; lam[a] = lam[c]; lam[c] = tl;                    \
    for (int i = 0; i < 3; ++i) {                                       \
      float tv = V[i * 3 + a]; V[i * 3 + a] = V[i * 3 + c];             \
      V[i * 3 + c] = tv;                                                \
    }                                                                   \
  } while (0)
  if (lam[0] < lam[1]) SWAPCOL(0, 1);
  if (lam[0] < lam[2]) SWAPCOL(0, 2);
  if (lam[1] < lam[2]) SWAPCOL(1, 2);
#undef SWAPCOL

  const float v0[3] = {V[0], V[3], V[6]};
  const float v1[3] = {V[1], V[4], V[7]};
  const float v2[3] = {V[2], V[5], V[8]};

  // U columns: u0 = normalize(H v0); u1 = GS(H v1); u2 = u0 x u1 (det U = +1)
  float u0[3], u1[3], u2[3];
#pragma unroll
  for (int i = 0; i < 3; ++i)
    u0[i] = H[i * 3 + 0] * v0[0] + H[i * 3 + 1] * v0[1] + H[i * 3 + 2] * v0[2];
  const float in0 =
      1.f / fmaxf(sqrtf(u0[0] * u0[0] + u0[1] * u0[1] + u0[2] * u0[2]), 1e-12f);
#pragma unroll
  for (int i = 0; i < 3; ++i) u0[i] *= in0;
#pragma unroll
  for (int i = 0; i < 3; ++i)
    u1[i] = H[i * 3 + 0] * v1[0] + H[i * 3 + 1] * v1[1] + H[i * 3 + 2] * v1[2];
  const float proj = u0[0] * u1[0] + u0[1] * u1[1] + u0[2] * u1[2];
#pragma unroll
  for (int i = 0; i < 3; ++i) u1[i] -= proj * u0[i];
  const float in1 =
      1.f / fmaxf(sqrtf(u1[0] * u1[0] + u1[1] * u1[1] + u1[2] * u1[2]), 1e-12f);
#pragma unroll
  for (int i = 0; i < 3; ++i) u1[i] *= in1;
  u2[0] = u0[1] * u1[2] - u0[2] * u1[1];
  u2[1] = u0[2] * u1[0] - u0[0] * u1[2];
  u2[2] = u0[0] * u1[1] - u0[1] * u1[0];

  // R = V diag(1,1,det(V)) U^T  (with det(U)=+1 construction above)
  const float dv = v0[0] * (v1[1] * v2[2] - v1[2] * v2[1]) -
                   v0[1] * (v1[0] * v2[2] - v1[2] * v2[0]) +
                   v0[2] * (v1[0] * v2[1] - v1[1] * v2[0]);
  const float e2 = (dv >= 0.f) ? 1.f : -1.f;

  float* o = ws2 + (size_t)b * 20;
#pragma unroll
  for (int i = 0; i < 3; ++i)
#pragma unroll
    for (int j = 0; j < 3; ++j)
      o[i * 3 + j] = v0[i] * u0[j] + v1[i] * u1[j] + e2 * v2[i] * u2[j];
  o[9] = muP[0];  o[10] = muP[1]; o[11] = muP[2];
  o[12] = muQ[0]; o[13] = muQ[1]; o[14] = muQ[2];

  const float nres = st[16];
  float d0 = 1.24f * cbrtf(fmaxf(nres - 15.f, 1e-3f)) - 1.8f;
  d0 = fmaxf(d0, 1e-3f);
  o[15] = 1.f / (d0 * d0);
  o[16] = cnt;
}

// ---------------------------------------------------------------- pass 2 ----
__global__ __launch_bounds__(256) void tm_pass2(
    const float* __restrict__ P, const float* __restrict__ Q,
    const unsigned char* __restrict__ CM, const float* __restrict__ ws2,
    float* __restrict__ ws3) {
  const int blk = blockIdx.x;
  const int b = blk >> 1;
  const int split = blk & 1;
  const int tid = threadIdx.x;
  const int lane = tid & 31;
  const int warp = tid >> 5;

  __shared__ float prm[17];
  __shared__ float wsum[8];
  if (tid < 17) prm[tid] = ws2[(size_t)b * 20 + tid];
  __syncthreads();

  const float R0 = prm[0], R1 = prm[1], R2 = prm[2];
  const float R3 = prm[3], R4 = prm[4], R5 = prm[5];
  const float R6 = prm[6], R7 = prm[7], R8 = prm[8];
  const float mP0 = prm[9], mP1 = prm[10], mP2 = prm[11];
  const float mQ0 = prm[12], mQ1 = prm[13], mQ2 = prm[14];
  const float inv = prm[15];

  const size_t pbase = (size_t)b * NN;
  const float4* __restrict__ Pv = (const float4*)(P + pbase * 3);
  const float4* __restrict__ Qv = (const float4*)(Q + pbase * 3);
  const unsigned int* __restrict__ CMv = (const unsigned int*)(CM + pbase);
  const int g0 = (split * PTS_PER_BLOCK) >> 2;

  float ssum = 0.f;
#pragma unroll
  for (int it = 0; it < 3; ++it) {
    const int p4 = g0 + it * 256 + tid;
    const float4 x0 = Pv[(size_t)p4 * 3 + 0];
    const float4 x1 = Pv[(size_t)p4 * 3 + 1];
    const float4 x2 = Pv[(size_t)p4 * 3 + 2];
    const float4 y0 = Qv[(size_t)p4 * 3 + 0];
    const float4 y1 = Qv[(size_t)p4 * 3 + 1];
    const float4 y2 = Qv[(size_t)p4 * 3 + 2];
    const unsigned int m4 = CMv[p4];

    const float px[4] = {x0.x, x0.w, x1.z, x2.y};
    const float py[4] = {x0.y, x1.x, x1.w, x2.z};
    const float pz[4] = {x0.z, x1.y, x2.x, x2.w};
    const float qx[4] = {y0.x, y0.w, y1.z, y2.y};
    const float qy[4] = {y0.y, y1.x, y1.w, y2.z};
    const float qz[4] = {y0.z, y1.y, y2.x, y2.w};

#pragma unroll
    for (int k = 0; k < 4; ++k) {
      const float w = (float)((m4 >> (8 * k)) & 1u);
      const float tx = px[k] - mP0, ty = py[k] - mP1, tz = pz[k] - mP2;
      const float ax = fmaf(R0, tx, fmaf(R1, ty, R2 * tz)) - (qx[k] - mQ0);
      const float ay = fmaf(R3, tx, fmaf(R4, ty, R5 * tz)) - (qy[k] - mQ1);
      const float az = fmaf(R6, tx, fmaf(R7, ty, R8 * tz)) - (qz[k] - mQ2);
      const float d2 = fmaf(ax, ax, fmaf(ay, ay, az * az));
      ssum += w / fmaf(d2, inv, 1.f);
    }
  }

#pragma unroll
  for (int m = 1; m <= 16; m <<= 1) ssum += __shfl_xor(ssum, m, 32);
  if (lane == 0) wsum[warp] = ssum;
  __syncthreads();
  if (tid == 0) {
    float t = 0.f;
#pragma unroll
    for (int w8 = 0; w8 < 8; ++w8) t += wsum[w8];
    ws3[blk] = t;
  }
}

// -------------------------------------------------------------- finalize ----
__global__ void tm_finalize(const float* __restrict__ ws2,
                            const float* __restrict__ ws3,
                            float* __restrict__ out) {
  const int b = blockIdx.x * blockDim.x + threadIdx.x;
  if (b >= BB) return;
  const float cnt = ws2[(size_t)b * 20 + 16];
  const float ssum = ws3[b * SPLIT + 0] + ws3[b * SPLIT + 1];
  out[b] = (cnt > 0.f) ? (ssum / fmaxf(cnt, 1.f)) : 0.f;
}

// ---------------------------------------------------------------- launch ----
extern "C" void kernel_launch(void* const* d_in, const int* in_sizes, int n_in,
                              void* d_out, int out_size, void* d_ws,
                              size_t ws_size, hipStream_t stream) {
  (void)in_sizes; (void)n_in; (void)out_size; (void)ws_size;
  const float* P = (const float*)d_in[0];
  const float* Q = (const float*)d_in[1];
  const unsigned char* CM = (const unsigned char*)d_in[2];  // [B,L,A] bool
  const unsigned char* RM = (const unsigned char*)d_in[3];  // [B,L] bool
  float* out = (float*)d_out;

  float* ws = (float*)d_ws;
  float* ws1 = ws;                    // [B*SPLIT][32] raw stats (+nres)
  float* ws2 = ws + (size_t)BB * SPLIT * 32;          // [B][20] R, mu, inv_d0sq, cnt
  float* ws3 = ws2 + (size_t)BB * 20;                 // [B*SPLIT] partial TM sums

  tm_pass1<<<BB * SPLIT, 256, 0, stream>>>(P, Q, CM, RM, ws1);
  tm_kabsch<<<(BB + 255) / 256, 256, 0, stream>>>(ws1, ws2);
  tm_pass2<<<BB * SPLIT, 256, 0, stream>>>(P, Q, CM, ws2, ws3);
  tm_finalize<<<(BB + 255) / 256, 256, 0, stream>>>(ws2, ws3, out);
}